// SpikeEncoder_68255620268196
// MI455X (gfx1250) — compile-verified
//
#include <hip/hip_runtime.h>

typedef __attribute__((ext_vector_type(2))) float    v2f;
typedef __attribute__((ext_vector_type(4))) float    v4f;
typedef __attribute__((ext_vector_type(8))) float    v8f;
typedef __attribute__((ext_vector_type(4))) unsigned u32x4;
typedef __attribute__((ext_vector_type(8))) int      i32x8;
typedef __attribute__((ext_vector_type(4))) int      i32x4;

#define BM 128
#define BN 128
#define BK 32
#define LDST (BK + 4)   // 36-float LDS row stride (bank-conflict-free); TDM pad reproduces it

// Low 32 bits of a flat LDS pointer = workgroup-relative LDS byte offset
// (the low 32 bits serve as the LDS address; HW adds per-wave LDS_BASE).
static __device__ __forceinline__ unsigned lds_off(const void* p) {
    return (unsigned)(uintptr_t)p;
}

// Per-lane async copy Memory -> LDS, 16B, GV addressing; tracked by ASYNCcnt.
static __device__ __forceinline__ void async_ld_b128(unsigned lds, unsigned long long ga) {
    asm volatile("global_load_async_to_lds_b128 %0, %1, off"
                 :: "v"(lds), "v"(ga) : "memory");
}
static __device__ __forceinline__ void wait_async0() {
    asm volatile("s_wait_asynccnt 0" ::: "memory");
}

// TDM: DMA a BN x BK f32 tile (row stride K elements) into LDS with hardware
// padding 32->36 dwords per row. One descriptor, wave-level, tracked by TENSORcnt.
static __device__ __forceinline__ void tdm_load_B(const float* gsrc, unsigned ldsDst, int K) {
    unsigned long long ga = (unsigned long long)(uintptr_t)gsrc;
    u32x4 g0;
    g0[0] = 1u;                                                // count=1, valid, gather off
    g0[1] = ldsDst;                                            // lds_addr (bytes)
    g0[2] = (unsigned)(ga & 0xFFFFFFFFu);                      // global_addr[31:0]
    g0[3] = (unsigned)((ga >> 32) & 0x01FFFFFFu) | (2u << 30); // global_addr[56:32] | type=2
    i32x8 g1;
    g1[0] = (int)((2u << 16)      // data_size = 4B
                | (1u << 20)      // pad_enable
                | (4u << 22)      // pad_interval: 32 dwords
                | (3u << 25));    // pad_amount: 4 dwords  (32+4 = 36-dword LDS rows)
    unsigned td0 = 1u << 30, td1 = 1u << 30;                   // huge tensor dims: no OOB clamp
    g1[1] = (int)((td0 & 0xFFFFu) << 16);
    g1[2] = (int)(((td0 >> 16) & 0xFFFFu) | ((td1 & 0xFFFFu) << 16));
    g1[3] = (int)(((td1 >> 16) & 0xFFFFu) | ((unsigned)BK << 16));   // tile_dim0 = 32
    g1[4] = (int)((unsigned)BN);                                     // tile_dim1 = 128, tile_dim2 = 0
    g1[5] = (int)(unsigned)K;                                        // tensor_dim0_stride (elements)
    g1[6] = 0;
    g1[7] = 0;
    i32x4 zz4; zz4[0] = zz4[1] = zz4[2] = zz4[3] = 0;
    i32x8 zz8;
    #pragma unroll
    for (int i = 0; i < 8; ++i) zz8[i] = 0;
    // 6-arg form (clang-23 / therock headers): (g0, g1, g2, g3, g4, cpol)
    __builtin_amdgcn_tensor_load_to_lds(g0, g1, zz4, zz4, zz8, 0);
}

// C[M,N] = rowgather(A, idx) @ Bm^T ;  out = (acc + bias_scale*bias[n]) * out_scale
// Double-buffered LDS: TDM streams B tiles, async-to-LDS streams gathered A rows,
// both overlapped with the f32 WMMA compute of the previous chunk.
__global__ __launch_bounds__(256)
void gemm_rowgather_f32wmma(const float* __restrict__ A, const int* __restrict__ idx,
                            const float* __restrict__ Bm, const float* __restrict__ bias,
                            float* __restrict__ C,
                            int M, int N, int K, int lda,
                            float bias_scale, float out_scale)
{
    __shared__ __align__(16) float sA[2 * BM * LDST];
    __shared__ __align__(16) float sB[2 * BN * LDST];
    __shared__ int sIdx[BM];

    const int tid  = threadIdx.x;
    const int lane = tid & 31;
    const int wid  = tid >> 5;
    const int wm   = wid & 3;    // 4 wave-rows  -> 32 output rows each
    const int wn   = wid >> 2;   // 2 wave-cols  -> 64 output cols each
    const int bm0  = blockIdx.x * BM;
    const int bn0  = blockIdx.y * BN;

    if (tid < BM) {
        int gr  = bm0 + tid;
        int src = 0;
        if (gr < M) src = idx ? idx[gr] : gr;
        sIdx[tid] = src;
    }
    __syncthreads();

    // stage A tile (BM x BK) for chunk k0 into buffer bufsel via async-to-LDS
    auto stageA = [&](int k0, int bufsel) {
        float* dst = &sA[bufsel * (BM * LDST)];
        #pragma unroll
        for (int it = 0; it < (BM * BK) / (256 * 4); ++it) {
            int q   = it * 256 + tid;     // one float4 per slot
            int row = q >> 3;             // 8 float4 per row
            int c4  = (q & 7) << 2;
            int gr  = bm0 + row;
            float* d = &dst[row * LDST + c4];
            if (gr < M) {
                const float* p = A + (size_t)sIdx[row] * lda + (k0 + c4);
                async_ld_b128(lds_off(d), (unsigned long long)(uintptr_t)p);
            } else {
                v4f z = {0.f, 0.f, 0.f, 0.f};
                *(v4f*)d = z;             // ragged rows: zero-fill (ds store)
            }
        }
    };

    v8f acc[2][4];
    #pragma unroll
    for (int mi = 0; mi < 2; ++mi)
        #pragma unroll
        for (int ni = 0; ni < 4; ++ni)
            #pragma unroll
            for (int v = 0; v < 8; ++v)
                acc[mi][ni][v] = 0.0f;

    // prologue: stage chunk 0 into buffer 0
    stageA(0, 0);
    if (wid == 0)
        tdm_load_B(Bm + (size_t)bn0 * K, lds_off(&sB[0]), K);
    wait_async0();
    __builtin_amdgcn_s_wait_tensorcnt(0);
    __syncthreads();

    for (int k0 = 0; k0 < K; k0 += BK) {
        const int buf = (k0 / BK) & 1;

        // issue next chunk's DMAs (overlaps with compute below)
        if (k0 + BK < K) {
            stageA(k0 + BK, buf ^ 1);
            if (wid == 0)
                tdm_load_B(Bm + (size_t)bn0 * K + (k0 + BK),
                           lds_off(&sB[(buf ^ 1) * (BN * LDST)]), K);
        }

        const float* cA = &sA[buf * (BM * LDST)];
        const float* cB = &sB[buf * (BN * LDST)];
        const int l16 = lane & 15;
        const int hi  = (lane >> 4) << 1;   // lanes 0-15: K=0,1 ; lanes 16-31: K=2,3
        #pragma unroll
        for (int kk = 0; kk < BK; kk += 4) {
            v2f aF[2], bF[4];
            #pragma unroll
            for (int mi = 0; mi < 2; ++mi)
                aF[mi] = *(const v2f*)&cA[(wm * 32 + mi * 16 + l16) * LDST + kk + hi];
            #pragma unroll
            for (int ni = 0; ni < 4; ++ni)
                bF[ni] = *(const v2f*)&cB[(wn * 64 + ni * 16 + l16) * LDST + kk + hi];
            #pragma unroll
            for (int mi = 0; mi < 2; ++mi)
                #pragma unroll
                for (int ni = 0; ni < 4; ++ni)
                    acc[mi][ni] = __builtin_amdgcn_wmma_f32_16x16x4_f32(
                        false, aF[mi], false, bF[ni],
                        (short)0, acc[mi][ni], false, false);
        }

        // retire the prefetch DMAs, then hand buffers over
        wait_async0();
        __builtin_amdgcn_s_wait_tensorcnt(0);
        __syncthreads();
    }

    // epilogue: D layout = lanes 0-15: (M=v, N=lane); lanes 16-31: (M=8+v, N=lane-16)
    const int l16 = lane & 15;
    const int mhi = (lane >> 4) << 3;
    #pragma unroll
    for (int mi = 0; mi < 2; ++mi) {
        #pragma unroll
        for (int ni = 0; ni < 4; ++ni) {
            int   gcol = bn0 + wn * 64 + ni * 16 + l16;
            float bv   = bias ? bias[gcol] : 0.f;
            #pragma unroll
            for (int v = 0; v < 8; ++v) {
                int grow = bm0 + wm * 32 + mi * 16 + mhi + v;
                if (grow < M)
                    C[(size_t)grow * N + gcol] =
                        (acc[mi][ni][v] + bias_scale * bv) * out_scale;
            }
        }
    }
}

// Per-neuron LIF dynamics collapsed to spike counts.
// mem2 accumulation is linear in spike -> only Sum_{s,t} spike[b,h] is needed.
__global__ __launch_bounds__(256)
void lif_count_kernel(const float* __restrict__ h1, float* __restrict__ cnt,
                      const float* __restrict__ thr_p, const float* __restrict__ leak_p,
                      int S, int BH)
{
    int g = blockIdx.x * blockDim.x + threadIdx.x;
    if (g >= BH) return;
    const float thr  = *thr_p;
    const float leak = *leak_p;
    float m1 = 0.f, c = 0.f;
    for (int s = 0; s < S; ++s) {
        float x = h1[(size_t)s * BH + g];
        #pragma unroll
        for (int t = 0; t < 10; ++t) {
            m1 = leak * m1 + x;
            if (m1 / thr - 1.0f > 0.f) {   // exact reference predicate
                c  += 1.f;                 // spike contributes 1.0 to count
                m1 -= thr;                 // soft reset
            }
        }
    }
    cnt[g] = c;
}

extern "C" void kernel_launch(void* const* d_in, const int* in_sizes, int n_in,
                              void* d_out, int out_size, void* d_ws, size_t ws_size,
                              hipStream_t stream) {
    const int*   input = (const int*)  d_in[0];   // [S,B] int32
    const float* emb   = (const float*)d_in[1];   // [V,E]
    const float* W1    = (const float*)d_in[2];   // [H,E]
    const float* b1    = (const float*)d_in[3];   // [H]
    const float* W2    = (const float*)d_in[4];   // [H,H]
    const float* b2    = (const float*)d_in[5];   // [H]
    const float* thr   = (const float*)d_in[6];
    const float* leak  = (const float*)d_in[7];

    const int S = 64, B = 64, E = 1024, H = 1024;
    const int M1 = S * B;                         // 4096 gathered rows

    float* h1  = (float*)d_ws;                    // [M1, H]   16 MB
    float* cnt = h1 + (size_t)M1 * H;             // [B*H]     256 KB

    // GEMM1: h1 = embedding[input] @ W1^T + b1
    dim3 g1(M1 / BM, H / BN);                     // 32 x 8
    gemm_rowgather_f32wmma<<<g1, 256, 0, stream>>>(
        emb, input, W1, b1, h1, M1, H, E, E, 1.0f, 1.0f);

    // LIF scan -> spike counts (640 sequential steps per neuron, parallel over 65536 neurons)
    lif_count_kernel<<<(B * H) / 256, 256, 0, stream>>>(h1, cnt, thr, leak, S, B * H);

    // GEMM2: out = (cnt @ W2^T + (S*T)*b2) / B
    dim3 g3((B + BM - 1) / BM, H / BN);           // 1 x 8
    gemm_rowgather_f32wmma<<<g3, 256, 0, stream>>>(
        cnt, nullptr, W2, b2, (float*)d_out, B, H, H, H, 640.0f, 1.0f / 64.0f);
}